// ConstGCN_214748365179
// MI455X (gfx1250) — compile-verified
//
#include <hip/hip_runtime.h>
#include <hip/hip_bf16.h>
#include <math.h>

#define NODES 8192   // B*S
#define DDIM  512
#define EDGES 65536  // NODES*DEG
#define DEG   8

#define KB      64   // K-chunk staged in LDS per step
#define PSTRIDE 72   // floats per LDS panel row (padding: conflict-free half-wave reads)

typedef float v2f __attribute__((ext_vector_type(2)));
typedef float v4f __attribute__((ext_vector_type(4)));
typedef float v8f __attribute__((ext_vector_type(8)));

// ---------------------------------------------------------------------------
// Kernel 1: X (8192x512) @ W (512x512) for three weight matrices.
// Block = 256 threads (8 waves) computing a 128x64 C panel; the shared 64x64
// B panel is staged into LDS with GLOBAL_LOAD_ASYNC_TO_LDS_B128 (ASYNCcnt),
// double-buffered so the copy of chunk c+1 overlaps WMMA on chunk c.
// Math: V_WMMA_F32_16X16X4_F32 (exact f32 on the matrix pipe).
// A layout (16x4 f32): lanes 0-15 -> M=lane, K={0,1}; lanes 16-31 -> K={2,3}.
// C layout: VGPR r -> M = r + 8*(lane>=16), N = lane&15.
// ---------------------------------------------------------------------------
__global__ __launch_bounds__(256)
void gcn_gemm_wmma(const float* __restrict__ X,
                   const float* __restrict__ W0,
                   const float* __restrict__ W1,
                   const float* __restrict__ W2,
                   float* __restrict__ P0,
                   float* __restrict__ P1,
                   float* __restrict__ P2)
{
    __shared__ __align__(16) float panel[2][KB * PSTRIDE];

    const int tid  = (int)threadIdx.x;
    const int lane = tid & 31;
    const int wave = tid >> 5;               // 0..7 -> 16-row slice of the panel
    const int mt   = (int)blockIdx.x;        // 128-row block
    const int nt   = (int)blockIdx.y;        // 64-col tile
    const int wsel = (int)blockIdx.z;        // which weight matrix

    const float* __restrict__ W = (wsel == 0) ? W0 : (wsel == 1) ? W1 : W2;
    float*       __restrict__ P = (wsel == 0) ? P0 : (wsel == 1) ? P1 : P2;

    const int m    = lane & 15;
    const int half = lane >> 4;              // 0: K={0,1}, 1: K={2,3}
    const int row  = mt * 128 + wave * 16 + m;
    const size_t rowOff = (size_t)row * DDIM;
    const int colBase = nt * 64;

    // B staging: one KB x 64 chunk = 1024 b128 segments; 256 threads x 4 each.
    // segment s: panel row = s>>4, col = (s&15)*4
    auto issue_chunk = [&](int k0, int buf) {
#pragma unroll
        for (int i = 0; i < 4; ++i) {
            const int s = tid + i * 256;
            const int r = s >> 4;
            const int c = (s & 15) * 4;
            const unsigned lds_addr =
                (unsigned)(uintptr_t)&panel[buf][r * PSTRIDE + c];
            const unsigned long long gaddr =
                (unsigned long long)(uintptr_t)(W + (size_t)(k0 + r) * DDIM + colBase + c);
            asm volatile("global_load_async_to_lds_b128 %0, %1, off"
                         :: "v"(lds_addr), "v"(gaddr) : "memory");
        }
    };

    v8f acc0 = {}, acc1 = {}, acc2 = {}, acc3 = {};

    issue_chunk(0, 0);

    for (int ch = 0; ch < DDIM / KB; ++ch) {
        const int buf = ch & 1;
        if (ch + 1 < DDIM / KB) {
            issue_chunk((ch + 1) * KB, buf ^ 1);
            // 8 outstanding per wave; <=4 left means current chunk has landed
            asm volatile("s_wait_asynccnt 0x4" ::: "memory");
        } else {
            asm volatile("s_wait_asynccnt 0x0" ::: "memory");
        }
        __syncthreads();

        const float* __restrict__ bp = &panel[buf][0];
        const float* __restrict__ ap = X + rowOff + ch * KB;
#pragma unroll 4
        for (int kk = 0; kk < KB; kk += 4) {
            const int ka = kk + half * 2;
            v2f a;
            a.x = ap[ka];
            a.y = ap[ka + 1];

            const float* __restrict__ bb = bp + ka * PSTRIDE + m;
            v2f b0, b1, b2, b3;
            b0.x = bb[0];   b0.y = bb[PSTRIDE];
            b1.x = bb[16];  b1.y = bb[16 + PSTRIDE];
            b2.x = bb[32];  b2.y = bb[32 + PSTRIDE];
            b3.x = bb[48];  b3.y = bb[48 + PSTRIDE];

            acc0 = __builtin_amdgcn_wmma_f32_16x16x4_f32(false, a, false, b0, (short)0, acc0, false, false);
            acc1 = __builtin_amdgcn_wmma_f32_16x16x4_f32(false, a, false, b1, (short)0, acc1, false, false);
            acc2 = __builtin_amdgcn_wmma_f32_16x16x4_f32(false, a, false, b2, (short)0, acc2, false, false);
            acc3 = __builtin_amdgcn_wmma_f32_16x16x4_f32(false, a, false, b3, (short)0, acc3, false, false);
        }
        __syncthreads();   // panel buf free for the chunk after next
    }

    const int rbase = half * 8;
#pragma unroll
    for (int r = 0; r < 8; ++r) {
        float* __restrict__ p =
            P + (size_t)(mt * 128 + wave * 16 + rbase + r) * DDIM + colBase + m;
        p[0]  = acc0[r];
        p[16] = acc1[r];
        p[32] = acc2[r];
        p[48] = acc3[r];
    }
}

// ---------------------------------------------------------------------------
// Kernel 2: gate GEMVs: g = x . v_gate for the three gate vectors.
// One wave per node; wave32 shuffle reduction.
// ---------------------------------------------------------------------------
__global__ __launch_bounds__(256)
void gcn_gates(const float* __restrict__ X,
               const float* __restrict__ Gin,
               const float* __restrict__ Gout,
               const float* __restrict__ Gloop,
               float* __restrict__ g_in,
               float* __restrict__ g_out,
               float* __restrict__ g_loop)
{
    const int tid  = (int)(blockIdx.x * blockDim.x + threadIdx.x);
    const int node = tid >> 5;
    const int lane = tid & 31;
    if (node >= NODES) return;

    const float* __restrict__ x = X + (size_t)node * DDIM;
    float s0 = 0.f, s1 = 0.f, s2 = 0.f;
#pragma unroll 4
    for (int j = lane; j < DDIM; j += 32) {
        const float xv = x[j];
        s0 += xv * Gin[j];
        s1 += xv * Gout[j];
        s2 += xv * Gloop[j];
    }
#pragma unroll
    for (int o = 16; o > 0; o >>= 1) {
        s0 += __shfl_xor(s0, o, 32);
        s1 += __shfl_xor(s1, o, 32);
        s2 += __shfl_xor(s2, o, 32);
    }
    if (lane == 0) {
        g_in[node]   = s0;
        g_out[node]  = s1;
        g_loop[node] = s2;
    }
}

// ---------------------------------------------------------------------------
// Kernel 3: per-node gather of 17 gated slots + LayerNorm + ReLU + residual.
// One wave per node; each lane owns 16 floats as 4 x float4 chunks.
// ---------------------------------------------------------------------------
__device__ __forceinline__ void gather_dir(int node, int lane,
                                           const int*   __restrict__ arc,
                                           const int*   __restrict__ lab,
                                           const float* __restrict__ mask,
                                           const float* __restrict__ P,
                                           const float* __restrict__ g,
                                           const float* __restrict__ bias,
                                           const float* __restrict__ bgate,
                                           v4f acc[4])
{
#pragma unroll
    for (int d = 0; d < DEG; ++d) {
        const int e   = node * DEG + d;
        const int idx = arc[e] * 512 + arc[EDGES + e];   // b*S + s, S == 512
        const int l   = lab[e];
        const float gate = g[idx] + bgate[l];
        const float prob = mask[e] * (1.0f / (1.0f + __expf(-gate)));
        const float* __restrict__ p  = P    + (size_t)idx * DDIM;
        const float* __restrict__ bb = bias + (size_t)l   * DDIM;
#pragma unroll
        for (int j = 0; j < 4; ++j) {
            const int c = j * 128 + lane * 4;
            const v4f pv = *(const v4f*)(p  + c);
            const v4f bv = *(const v4f*)(bb + c);
            acc[j] += prob * (pv + bv);
        }
    }
}

__global__ __launch_bounds__(256)
void gcn_aggregate(const float* __restrict__ src,
                   const int*   __restrict__ arc_in,  const int* __restrict__ lab_in,
                   const int*   __restrict__ arc_out, const int* __restrict__ lab_out,
                   const float* __restrict__ mask_in, const float* __restrict__ mask_out,
                   const float* __restrict__ mask_loop,
                   const float* __restrict__ sent_mask,
                   const float* __restrict__ b_in,  const float* __restrict__ b_in_gate,
                   const float* __restrict__ b_out, const float* __restrict__ b_out_gate,
                   const float* __restrict__ gamma, const float* __restrict__ beta,
                   const float* __restrict__ P_in,  const float* __restrict__ P_out,
                   const float* __restrict__ P_loop,
                   const float* __restrict__ g_in,  const float* __restrict__ g_out,
                   const float* __restrict__ g_loop,
                   float* __restrict__ out)
{
    const int tid  = (int)(blockIdx.x * blockDim.x + threadIdx.x);
    const int node = tid >> 5;
    const int lane = tid & 31;
    if (node >= NODES) return;

    v4f acc[4] = {};

    gather_dir(node, lane, arc_in,  lab_in,  mask_in,  P_in,  g_in,  b_in,  b_in_gate,  acc);
    gather_dir(node, lane, arc_out, lab_out, mask_out, P_out, g_out, b_out, b_out_gate, acc);

    // self-loop slot (no label bias)
    {
        const float gate = g_loop[node];
        const float prob = mask_loop[node] * (1.0f / (1.0f + __expf(-gate)));
        const float* __restrict__ p = P_loop + (size_t)node * DDIM;
#pragma unroll
        for (int j = 0; j < 4; ++j) {
            const int c = j * 128 + lane * 4;
            acc[j] += prob * (*(const v4f*)(p + c));
        }
    }

    // LayerNorm over 512 elems: wave32 reduction of sum and sum-of-squares
    float s1 = 0.f, s2 = 0.f;
#pragma unroll
    for (int j = 0; j < 4; ++j) {
        s1 += acc[j].x + acc[j].y + acc[j].z + acc[j].w;
        s2 += acc[j].x * acc[j].x + acc[j].y * acc[j].y +
              acc[j].z * acc[j].z + acc[j].w * acc[j].w;
    }
#pragma unroll
    for (int o = 16; o > 0; o >>= 1) {
        s1 += __shfl_xor(s1, o, 32);
        s2 += __shfl_xor(s2, o, 32);
    }
    const float mu  = s1 * (1.0f / (float)DDIM);
    const float var = s2 * (1.0f / (float)DDIM) - mu * mu;
    const float rs  = rsqrtf(var + 1e-5f);
    const float sm  = sent_mask[node];

#pragma unroll
    for (int j = 0; j < 4; ++j) {
        const int c = j * 128 + lane * 4;
        const v4f gm = *(const v4f*)(gamma + c);
        const v4f bt = *(const v4f*)(beta + c);
        const v4f sv = *(const v4f*)(src + (size_t)node * DDIM + c);
        v4f y = (acc[j] - mu) * rs * gm + bt;
        y *= sm;                                   // mask before ReLU
        y.x = fmaxf(y.x, 0.f);
        y.y = fmaxf(y.y, 0.f);
        y.z = fmaxf(y.z, 0.f);
        y.w = fmaxf(y.w, 0.f);
        y = y * sm + sv;                           // mask again, then residual
        *(v4f*)(out + (size_t)node * DDIM + c) = y;
    }
}

// ---------------------------------------------------------------------------
extern "C" void kernel_launch(void* const* d_in, const int* in_sizes, int n_in,
                              void* d_out, int out_size, void* d_ws, size_t ws_size,
                              hipStream_t stream)
{
    const float* src        = (const float*)d_in[0];
    const int*   arc_in     = (const int*)d_in[1];
    const int*   lab_in     = (const int*)d_in[2];
    const int*   arc_out    = (const int*)d_in[3];
    const int*   lab_out    = (const int*)d_in[4];
    const float* mask_in    = (const float*)d_in[5];
    const float* mask_out   = (const float*)d_in[6];
    const float* mask_loop  = (const float*)d_in[7];
    const float* sent_mask  = (const float*)d_in[8];
    const float* V_in       = (const float*)d_in[9];
    const float* b_in       = (const float*)d_in[10];
    const float* V_in_gate  = (const float*)d_in[11];
    const float* b_in_gate  = (const float*)d_in[12];
    const float* V_out      = (const float*)d_in[13];
    const float* b_out      = (const float*)d_in[14];
    const float* V_out_gate = (const float*)d_in[15];
    const float* b_out_gate = (const float*)d_in[16];
    const float* W_loop     = (const float*)d_in[17];
    const float* W_loop_g   = (const float*)d_in[18];
    const float* ln_gamma   = (const float*)d_in[19];
    const float* ln_beta    = (const float*)d_in[20];

    float* P_in   = (float*)d_ws;
    float* P_out  = P_in   + (size_t)NODES * DDIM;
    float* P_loop = P_out  + (size_t)NODES * DDIM;
    float* g_in   = P_loop + (size_t)NODES * DDIM;
    float* g_out  = g_in   + NODES;
    float* g_loop = g_out  + NODES;

    dim3 gemm_grid(NODES / 128, DDIM / 64, 3);
    gcn_gemm_wmma<<<gemm_grid, 256, 0, stream>>>(src, V_in, V_out, W_loop,
                                                 P_in, P_out, P_loop);

    gcn_gates<<<NODES / 8, 256, 0, stream>>>(src, V_in_gate, V_out_gate, W_loop_g,
                                             g_in, g_out, g_loop);

    gcn_aggregate<<<NODES / 8, 256, 0, stream>>>(
        src, arc_in, lab_in, arc_out, lab_out,
        mask_in, mask_out, mask_loop, sent_mask,
        b_in, b_in_gate, b_out, b_out_gate,
        ln_gamma, ln_beta,
        P_in, P_out, P_loop, g_in, g_out, g_loop,
        (float*)d_out);
}